// attentionLayer_1675037246228
// MI455X (gfx1250) — compile-verified
//
#include <hip/hip_runtime.h>
#include <math.h>

// CDNA5 f32 WMMA fragments: A 16x4 / B 4x16 = 2 VGPRs, C/D 16x16 = 8 VGPRs
typedef __attribute__((ext_vector_type(2))) float v2f;
typedef __attribute__((ext_vector_type(8))) float v8f;

#define B_ 32
#define M_ 16
#define S_ 64
#define Q_ 48
#define D_ 256

#define U_STRIDE 257   // 257 % 64 == 1 -> conflict-free column reads of U^T
#define S_STRIDE 49    // gcd(49,64)==1 -> conflict-free A-frag reads of `a`

__global__ __launch_bounds__(384)
void bidaf_attention_fused(const float* __restrict__ h_g,
                           const float* __restrict__ u_g,
                           const float* __restrict__ W_g,
                           const float* __restrict__ b_g,
                           float* __restrict__ out)
{
    __shared__ float U_lds[Q_ * U_STRIDE];   // 48 x 256 (padded)  ~48.2 KB
    __shared__ float s_lds[S_ * S_STRIDE];   // 64 x 48  (padded)  ~12.3 KB
    __shared__ float whu[D_];                // w_hu
    __shared__ float hw[S_];                 // h[i,:]·w_h
    __shared__ float uw[Q_];                 // u[j,:]·w_u
    __shared__ float mrow[S_];               // max_j s[i,j]
    __shared__ float p_lds[S_];              // softmax_i(mrow)
    __shared__ float hatt[D_];               // p^T H

    const int tid  = threadIdx.x;
    const int lane = tid & 31;
    const int wave = tid >> 5;               // 0..11
    const int bm   = blockIdx.x;             // b*M + m
    const int bb   = bm / M_;

    const float* hB   = h_g + (size_t)bm * (S_ * D_);
    const float* uB   = u_g + (size_t)bb * (Q_ * D_);
    float*       outB = out + (size_t)bm * (S_ * 4 * D_);

    // ---- stage U[b] into LDS (padded rows); stage w_hu ----
    for (int idx = tid; idx < Q_ * (D_ / 4); idx += 384) {
        const int row = idx >> 6;            // 64 float4 per row
        const int c4  = (idx & 63) << 2;
        const float4 v = *(const float4*)(uB + row * D_ + c4);
        float* dst = &U_lds[row * U_STRIDE + c4];
        dst[0] = v.x; dst[1] = v.y; dst[2] = v.z; dst[3] = v.w;
    }
    if (tid < D_) whu[tid] = W_g[2 * D_ + tid];
    __syncthreads();

    // ---- rank-1 terms via wave-parallel coalesced reductions ----
    // waves 0..7: hw[i] for rows 8w..8w+7 (lanes read contiguous 128B slices);
    // waves 8..11: uw[j] for 12 rows each from LDS. 5-step shfl_xor tree reduce.
    if (wave < 8) {
        const int i0 = wave * 8;
        for (int r = 0; r < 8; ++r) {
            const int i = i0 + r;
            float acc = 0.f;
            #pragma unroll
            for (int t = 0; t < 8; ++t) {
                const int d = lane + 32 * t;
                acc += hB[i * D_ + d] * W_g[d];
            }
            #pragma unroll
            for (int off = 16; off > 0; off >>= 1) acc += __shfl_xor(acc, off);
            if (lane == 0) hw[i] = acc;
        }
    } else {
        const int j0 = (wave - 8) * 12;
        for (int r = 0; r < 12; ++r) {
            const int j = j0 + r;
            float acc = 0.f;
            #pragma unroll
            for (int t = 0; t < 8; ++t) {
                const int d = lane + 32 * t;
                acc += U_lds[j * U_STRIDE + d] * W_g[D_ + d];
            }
            #pragma unroll
            for (int off = 16; off > 0; off >>= 1) acc += __shfl_xor(acc, off);
            if (lane == 0) uw[j] = acc;
        }
    }
    __syncthreads();

    const float bias = b_g[0];
    const int   kh   = (lane >> 4) << 1;     // k-pair select per lane-half

    // ---- GEMM1: s = (H ⊙ w_hu) @ U^T   (64x256 @ 256x48), 12 waves = 4x3 tiles ----
    {
        const int mt = wave / 3;                      // 0..3
        const int nt = wave % 3;                      // 0..2
        const int mA = mt * 16 + (lane & 15);
        const int jB = nt * 16 + (lane & 15);
        const float* hrow = hB + mA * D_;
        v8f c = {};
        for (int k0 = 0; k0 < D_; k0 += 4) {
            const int kk = k0 + kh;
            const float2 hv = *(const float2*)(hrow + kk);
            v2f af, bf;
            af.x = hv.x * whu[kk];
            af.y = hv.y * whu[kk + 1];
            bf.x = U_lds[jB * U_STRIDE + kk];         // U^T column = U row
            bf.y = U_lds[jB * U_STRIDE + kk + 1];
            c = __builtin_amdgcn_wmma_f32_16x16x4_f32(false, af, false, bf,
                                                      (short)0, c, false, false);
        }
        const int i0 = mt * 16 + ((lane >> 4) << 3);
        const int j  = nt * 16 + (lane & 15);
        #pragma unroll
        for (int r = 0; r < 8; ++r) {
            const int i = i0 + r;
            s_lds[i * S_STRIDE + j] = c[r] + hw[i] + uw[j] + bias;
        }
    }
    __syncthreads();

    // ---- softmax over j (in place), record row maxes ----
    if (tid < S_) {
        float* row = &s_lds[tid * S_STRIDE];
        float mx = row[0];
        for (int j = 1; j < Q_; ++j) mx = fmaxf(mx, row[j]);
        float sum = 0.f;
        for (int j = 0; j < Q_; ++j) { float e = __expf(row[j] - mx); row[j] = e; sum += e; }
        const float inv = 1.f / sum;
        for (int j = 0; j < Q_; ++j) row[j] *= inv;
        mrow[tid] = mx;
    }
    __syncthreads();

    // ---- p = softmax_i(mrow)  (64 values; redundant per-thread scan is cheap) ----
    if (tid < S_) {
        float pm = mrow[0];
        for (int i = 1; i < S_; ++i) pm = fmaxf(pm, mrow[i]);
        float ps = 0.f;
        for (int i = 0; i < S_; ++i) ps += __expf(mrow[i] - pm);
        p_lds[tid] = __expf(mrow[tid] - pm) / ps;
    }
    __syncthreads();

    // ---- h_att[d] = sum_i p[i] * H[i,d]  (coalesced over d) ----
    if (tid < D_) {
        float acc = 0.f;
        for (int i = 0; i < S_; ++i) acc += p_lds[i] * hB[i * D_ + tid];
        hatt[tid] = acc;
    }
    __syncthreads();

    // ---- GEMM2: u_att = a @ U (64x48 @ 48x256), fused fully-coalesced epilogue ----
    // Each wave computes a 16x32 tile (two 16x16 WMMA accumulators side by side);
    // a half-wave shfl_xor(16) reshapes C so every global store is one contiguous
    // 128B segment. Output is streamed with non-temporal hints (write-once data
    // must not evict the L2-resident H/U working set).
    for (int tile = wave; tile < 32; tile += 12) {
        const int mt    = tile >> 3;                  // 0..3
        const int ntp   = tile & 7;                   // 0..7  (32-wide column panel)
        const int dbase = ntp * 32;
        const int am    = (mt * 16 + (lane & 15)) * S_STRIDE;
        const int dA    = dbase + (lane & 15);
        const int dB    = dA + 16;
        v8f cA = {}, cB = {};
        for (int k0 = 0; k0 < Q_; k0 += 4) {
            const int kk = k0 + kh;
            v2f af, bfA, bfB;
            af.x  = s_lds[am + kk];
            af.y  = s_lds[am + kk + 1];
            bfA.x = U_lds[kk * U_STRIDE + dA];
            bfA.y = U_lds[(kk + 1) * U_STRIDE + dA];
            bfB.x = U_lds[kk * U_STRIDE + dB];
            bfB.y = U_lds[(kk + 1) * U_STRIDE + dB];
            cA = __builtin_amdgcn_wmma_f32_16x16x4_f32(false, af, false, bfA,
                                                       (short)0, cA, false, false);
            cB = __builtin_amdgcn_wmma_f32_16x16x4_f32(false, af, false, bfB,
                                                       (short)0, cB, false, false);
        }
        const int   d   = dbase + lane;               // contiguous over the wave
        const float ha  = hatt[d];
        const bool  lo  = (lane < 16);
        #pragma unroll
        for (int r = 0; r < 8; ++r) {
            const int iR  = mt * 16 + r;              // row r
            const int iR8 = iR + 8;                   // row r+8
            const float xA = __shfl_xor(cA[r], 16);   // half-wave swap
            const float xB = __shfl_xor(cB[r], 16);
            const float uR  = lo ? cA[r] : xB;        // u_att[iR ][d]
            const float uR8 = lo ? xA    : cB[r];     // u_att[iR8][d]
            const float hvR  = hB[iR  * D_ + d];      // coalesced 128B read
            const float hvR8 = hB[iR8 * D_ + d];
            float* oR  = outB + iR  * (4 * D_);
            float* oR8 = outB + iR8 * (4 * D_);
            __builtin_nontemporal_store(hvR,       oR + d);           // h
            __builtin_nontemporal_store(uR,        oR + D_ + d);      // u_att
            __builtin_nontemporal_store(hvR * uR,  oR + 2 * D_ + d);  // h ⊙ u_att
            __builtin_nontemporal_store(hvR * ha,  oR + 3 * D_ + d);  // h ⊙ h_att
            __builtin_nontemporal_store(hvR8,      oR8 + d);
            __builtin_nontemporal_store(uR8,       oR8 + D_ + d);
            __builtin_nontemporal_store(hvR8 * uR8, oR8 + 2 * D_ + d);
            __builtin_nontemporal_store(hvR8 * ha, oR8 + 3 * D_ + d);
        }
    }
}

extern "C" void kernel_launch(void* const* d_in, const int* in_sizes, int n_in,
                              void* d_out, int out_size, void* d_ws, size_t ws_size,
                              hipStream_t stream) {
    const float* h = (const float*)d_in[0];   // (B,M,S,D) f32
    const float* u = (const float*)d_in[1];   // (B,Q,D)   f32
    const float* W = (const float*)d_in[2];   // (3D,)     f32
    const float* b = (const float*)d_in[3];   // (1,)      f32
    float* out = (float*)d_out;               // (B,M,S,4D) f32

    dim3 grid(B_ * M_);                       // 512 blocks, one per (b,m)
    dim3 block(384);                          // 12 wave32s
    hipLaunchKernelGGL(bidaf_attention_fused, grid, block, 0, stream,
                       h, u, W, b, out);
}